// MultiheadAttention_43061342109988
// MI455X (gfx1250) — compile-verified
//
#include <hip/hip_runtime.h>
#include <hip/hip_bf16.h>
#include <math.h>

#define NN 80000
#define EE 640000
#define DD 128
#define HH 4
#define DKK 32

typedef float v2f __attribute__((ext_vector_type(2)));
typedef float v8f __attribute__((ext_vector_type(8)));

static __device__ __forceinline__ unsigned int enc_f32(float f) {
    unsigned int b = __float_as_uint(f);
    return (b & 0x80000000u) ? ~b : (b | 0x80000000u);
}
static __device__ __forceinline__ float dec_f32(unsigned int e) {
    unsigned int b = (e & 0x80000000u) ? (e & 0x7FFFFFFFu) : ~e;
    return __uint_as_float(b);
}

// C[m, n] = sum_d A[m, d] * W[n, d]   (C = A @ W^T)
// A: [M, 128] row-major, W: [128, 128] row-major, M % 16 == 0.
// One wave per 16x16 output tile; K accumulated with V_WMMA_F32_16X16X4_F32 (32 steps).
__global__ __launch_bounds__(256) void gemm_AWt(const float* __restrict__ A,
                                                const float* __restrict__ W,
                                                float* __restrict__ C,
                                                int M) {
    const int lane = threadIdx.x & 31;
    const int wave = threadIdx.x >> 5;
    const int tile = blockIdx.x * 8 + wave;           // wave-uniform
    const int colTiles = DD / 16;                     // 8
    const int mTiles = M >> 4;
    if (tile >= mTiles * colTiles) return;            // whole wave exits together
    const int mt = tile / colTiles;
    const int nt = tile % colTiles;

    const int half = lane >> 4;                       // 0: K={0,1}, 1: K={2,3} per step
    const int l15  = lane & 15;

    // A 16x4 fp32 layout: lanes 0-15 rows M=0..15 with (K0,K1); lanes 16-31 same rows with (K2,K3)
    const float* aPtr = A + (size_t)(mt * 16 + l15) * DD + half * 2;
    // B 4x16 fp32 layout (B[k][n] = W[n][k]): lanes 0-15 cols N=0..15 with (K0,K1); lanes 16-31 (K2,K3)
    const float* wPtr = W + (size_t)(nt * 16 + l15) * DD + half * 2;

    v8f c = {};
#pragma unroll
    for (int k0 = 0; k0 < DD; k0 += 4) {
        v2f a = *(const v2f*)(aPtr + k0);
        v2f b = *(const v2f*)(wPtr + k0);
        c = __builtin_amdgcn_wmma_f32_16x16x4_f32(false, a, false, b,
                                                  (short)0, c, false, false);
    }

    // C/D 16x16 fp32 layout: vgpr r, lane l -> M = r + 8*(l>>4), N = l&15
    float* out = C + (size_t)(mt * 16 + half * 8) * DD + nt * 16 + l15;
#pragma unroll
    for (int r = 0; r < 8; ++r)
        out[(size_t)r * DD] = c[r];
}

// One wave per edge. Lanes: head h = lane>>3, sub-lane j = lane&7, each lane dots 4 dk.
__global__ __launch_bounds__(256) void edge_scores(const float* __restrict__ q,
                                                   const float* __restrict__ k,
                                                   const int* __restrict__ src,
                                                   const int* __restrict__ dst,
                                                   float* __restrict__ score,
                                                   unsigned int* __restrict__ m_enc) {
    const int lane = threadIdx.x & 31;
    const int wave = threadIdx.x >> 5;
    const int e = blockIdx.x * 8 + wave;
    if (e >= EE) return;
    const int s = src[e], dn = dst[e];
    const int h = lane >> 3;
    const int j = lane & 7;
    const float4 kv = *(const float4*)(k + (size_t)s  * DD + h * DKK + j * 4);
    const float4 qv = *(const float4*)(q + (size_t)dn * DD + h * DKK + j * 4);
    float dot = kv.x * qv.x + kv.y * qv.y + kv.z * qv.z + kv.w * qv.w;
    dot += __shfl_xor(dot, 1);
    dot += __shfl_xor(dot, 2);
    dot += __shfl_xor(dot, 4);
    if (j == 0) {
        const float sc = dot * 0.17677669529663687f; // 1/sqrt(32)
        score[(size_t)e * HH + h] = sc;
        atomicMax(&m_enc[(size_t)dn * HH + h], enc_f32(sc));
    }
}

// Fused softmax-numerator + value aggregation: one edge pass.
__global__ __launch_bounds__(256) void edge_aggregate(const float* __restrict__ v,
                                                      const int* __restrict__ src,
                                                      const int* __restrict__ dst,
                                                      const float* __restrict__ score,
                                                      const unsigned int* __restrict__ m_enc,
                                                      float* __restrict__ den,
                                                      float* __restrict__ out_pre) {
    const int lane = threadIdx.x & 31;
    const int wave = threadIdx.x >> 5;
    const int e = blockIdx.x * 8 + wave;
    if (e >= EE) return;
    const int s = src[e], dn = dst[e];
    const int h = lane >> 3;
    const int j = lane & 7;
    const float sc = score[(size_t)e * HH + h];
    const float m  = dec_f32(m_enc[(size_t)dn * HH + h]);
    const float att = __expf(sc - m);
    if (j == 0) atomicAdd(&den[(size_t)dn * HH + h], att);
    const float4 vv = *(const float4*)(v + (size_t)s * DD + h * DKK + j * 4);
    float* o = out_pre + (size_t)dn * DD + h * DKK + j * 4;
    atomicAdd(o + 0, att * vv.x);
    atomicAdd(o + 1, att * vv.y);
    atomicAdd(o + 2, att * vv.z);
    atomicAdd(o + 3, att * vv.w);
}

// out_pre[n, h, :] /= den[n, h]; zero-in-degree nodes -> 0 (matches segment_sum default).
__global__ __launch_bounds__(256) void normalize_agg(float* __restrict__ out_pre,
                                                     const float* __restrict__ den) {
    const int idx = blockIdx.x * blockDim.x + threadIdx.x;
    if (idx >= NN * DD) return;
    const int n = idx >> 7;
    const int h = (idx >> 5) & 3;
    const float dv = den[n * HH + h];
    out_pre[idx] = (dv > 0.0f) ? (out_pre[idx] / dv) : 0.0f;
}

extern "C" void kernel_launch(void* const* d_in, const int* in_sizes, int n_in,
                              void* d_out, int out_size, void* d_ws, size_t ws_size,
                              hipStream_t stream) {
    const float* x  = (const float*)d_in[0];
    const float* WQ = (const float*)d_in[1];
    const float* WK = (const float*)d_in[2];
    const float* WV = (const float*)d_in[3];
    const float* WO = (const float*)d_in[4];
    const int* src  = (const int*)d_in[5];
    const int* dst  = (const int*)d_in[6];

    // workspace layout (floats / uints)
    float* q        = (float*)d_ws;                         // N*D
    float* k        = q + (size_t)NN * DD;                  // N*D
    float* v        = k + (size_t)NN * DD;                  // N*D
    float* score    = v + (size_t)NN * DD;                  // E*H
    unsigned int* m_enc = (unsigned int*)(score + (size_t)EE * HH); // N*H
    float* den      = (float*)(m_enc + (size_t)NN * HH);    // N*H
    float* out_pre  = den + (size_t)NN * HH;                // N*D

    hipMemsetAsync(m_enc, 0, sizeof(unsigned int) * (size_t)NN * HH, stream); // 0 < enc(any finite)
    hipMemsetAsync(den, 0, sizeof(float) * (size_t)NN * HH, stream);
    hipMemsetAsync(out_pre, 0, sizeof(float) * (size_t)NN * DD, stream);

    const dim3 blk(256);
    const int gemmBlocks = (NN / 16) * (DD / 16) / 8;   // 5000 blocks, 8 waves each
    const int edgeBlocks = EE / 8;                      // 80000 blocks, 1 wave/edge

    gemm_AWt<<<gemmBlocks, blk, 0, stream>>>(x, WQ, q, NN);
    gemm_AWt<<<gemmBlocks, blk, 0, stream>>>(x, WK, k, NN);
    gemm_AWt<<<gemmBlocks, blk, 0, stream>>>(x, WV, v, NN);

    edge_scores<<<edgeBlocks, blk, 0, stream>>>(q, k, src, dst, score, m_enc);
    edge_aggregate<<<edgeBlocks, blk, 0, stream>>>(v, src, dst, score, m_enc, den, out_pre);

    normalize_agg<<<(NN * DD + 255) / 256, blk, 0, stream>>>(out_pre, den);

    gemm_AWt<<<gemmBlocks, blk, 0, stream>>>(out_pre, WO, (float*)d_out, NN);
}